// AgMixPooler_1206_3650722201918
// MI455X (gfx1250) — compile-verified
//
#include <hip/hip_runtime.h>
#include <hip/hip_bf16.h>
#include <math.h>

typedef __attribute__((ext_vector_type(2))) float v2f;
typedef __attribute__((ext_vector_type(8))) float v8f;

#define Bb   8
#define Tt   8192
#define Ee   512
#define Dd   64
#define KK   1024
#define WINW 7

// low 32 bits of a generic pointer to __shared__ == LDS byte offset (aperture model)
__device__ __forceinline__ unsigned lds_off(const void* p) {
  return (unsigned)(unsigned long long)p;
}
__device__ __forceinline__ void async_load_lds_b128(unsigned lds, const void* gptr) {
  asm volatile("global_load_async_to_lds_b128 %0, %1, off"
               :: "v"(lds), "v"((unsigned long long)gptr) : "memory");
}
__device__ __forceinline__ void wait_asynccnt0() {
  asm volatile("s_wait_asynccnt 0" ::: "memory");
}

// ---------------------------------------------------------------------------
// Kernel 1: h = LN(gelu(X @ W1 + b1)) via V_WMMA_F32_16X16X4_F32
// block = 256 threads (8 waves), covers 128 rows of X; grid = 65536/128 = 512
// Xc staged via GLOBAL_LOAD_ASYNC_TO_LDS_B128 (ASYNCcnt path);
// Wc staged TRANSPOSED (Wc[n][k]) so B fragments are single contiguous b64 reads.
// ---------------------------------------------------------------------------
#define TILE_M 128
#define KCHUNK 64
#define XS 68   // padded LDS stride (bank-conflict-free fragment reads)

__global__ __launch_bounds__(256) void gemm_gelu_ln_kernel(
    const float* __restrict__ X, const float* __restrict__ W,
    const float* __restrict__ b1, const float* __restrict__ ln_g,
    const float* __restrict__ ln_b, float* __restrict__ h_out)
{
  __shared__ float Xc[TILE_M][XS];
  __shared__ float Wc[Dd][XS];        // transposed: [n][k], stride 68

  const int tid  = threadIdx.x;
  const int wave = tid >> 5;
  const int lane = tid & 31;
  const int half = lane >> 4;         // 0: lanes 0-15, 1: lanes 16-31
  const int mrow = lane & 15;
  const long rowBase = (long)blockIdx.x * TILE_M;

  v8f acc[4] = {};

  for (int kc = 0; kc < Ee; kc += KCHUNK) {
    // --- stage X tile: 128 rows x 64 k, async copy straight into LDS ---
    #pragma unroll
    for (int i = 0; i < 8; ++i) {
      int fi = tid + i * 256;          // 0..2047 float4 slots
      int r  = fi >> 4;
      int c4 = (fi & 15) << 2;
      async_load_lds_b128(lds_off(&Xc[r][c4]),
                          X + (rowBase + r) * Ee + kc + c4);
    }
    // --- stage W chunk transposed: Wc[n][k]; coalesced column gathers ---
    #pragma unroll
    for (int i = 0; i < 4; ++i) {
      int fi = tid + i * 256;          // 0..1023
      int n  = fi & 63;                // consecutive lanes -> consecutive n
      int k4 = (fi >> 6) << 2;         // 0,4,...,60
      float4 v;
      v.x = W[(long)(kc + k4 + 0) * Dd + n];
      v.y = W[(long)(kc + k4 + 1) * Dd + n];
      v.z = W[(long)(kc + k4 + 2) * Dd + n];
      v.w = W[(long)(kc + k4 + 3) * Dd + n];
      *reinterpret_cast<float4*>(&Wc[n][k4]) = v;
    }
    wait_asynccnt0();
    __syncthreads();

    const int rw = wave * 16 + mrow;   // this lane's X row within tile
    #pragma unroll
    for (int k0 = 0; k0 < KCHUNK; k0 += 4) {
      // A 16x4: lanes 0-15 hold K={k0,k0+1}, lanes 16-31 K={k0+2,k0+3}
      v2f a = *reinterpret_cast<const v2f*>(&Xc[rw][k0 + 2 * half]);
      #pragma unroll
      for (int nt = 0; nt < 4; ++nt) {
        // B 4x16 mirrors A's lane split; transposed Wc makes it contiguous
        v2f b = *reinterpret_cast<const v2f*>(&Wc[nt * 16 + mrow][k0 + 2 * half]);
        acc[nt] = __builtin_amdgcn_wmma_f32_16x16x4_f32(
            false, a, false, b, (short)0, acc[nt], false, false);
      }
    }
    __syncthreads();
  }

  // epilogue: +b1, exact GELU, LayerNorm over D=64, store
  float bb[4], gg[4], be[4];
  #pragma unroll
  for (int nt = 0; nt < 4; ++nt) {
    int n = nt * 16 + mrow;
    bb[nt] = b1[n]; gg[nt] = ln_g[n]; be[nt] = ln_b[n];
  }

  #pragma unroll
  for (int i = 0; i < 8; ++i) {        // C VGPR i: row = i + 8*half
    float s = 0.f, s2 = 0.f;
    #pragma unroll
    for (int nt = 0; nt < 4; ++nt) {
      float x = acc[nt][i] + bb[nt];
      x = 0.5f * x * (1.0f + erff(x * 0.70710678118654752f));
      acc[nt][i] = x;
      s += x; s2 += x * x;
    }
    #pragma unroll
    for (int off = 1; off < 16; off <<= 1) {   // width-16 keeps halves apart
      s  += __shfl_xor(s,  off, 16);
      s2 += __shfl_xor(s2, off, 16);
    }
    float mu   = s * (1.0f / 64.0f);
    float var  = s2 * (1.0f / 64.0f) - mu * mu;
    float rstd = rsqrtf(var + 1e-5f);
    long gr = rowBase + wave * 16 + i + 8 * half;
    #pragma unroll
    for (int nt = 0; nt < 4; ++nt) {
      float x = (acc[nt][i] - mu) * rstd * gg[nt] + be[nt];
      h_out[gr * Dd + nt * 16 + mrow] = x;
    }
  }
}

// ---------------------------------------------------------------------------
// Kernel 2: w_conv (7-tap over T, summed over D) + ssf + gated tanh + mask
// block = 128 threads = 128 t's; grid = B * T/128 = 512
// h tile staged with async-to-LDS; out-of-range rows zero-filled via ds stores.
// ---------------------------------------------------------------------------
#define CT 128
__global__ __launch_bounds__(128) void conv_gate_kernel(
    const float* __restrict__ h, const float* __restrict__ conv_w,
    const float* __restrict__ conv_b, const float* __restrict__ ssf_x,
    const float* __restrict__ ssf_w, const float* __restrict__ ssf_b,
    const float* __restrict__ gate_logit, const unsigned char* __restrict__ pmask,
    float* __restrict__ a_out)
{
  __shared__ float hb[CT + 6][68];
  __shared__ float cw[WINW][64];

  const int tid   = threadIdx.x;
  const int b     = blockIdx.x >> 6;
  const int chunk = blockIdx.x & 63;
  const long t0   = (long)chunk * CT;
  const long base = (long)b * Tt;

  for (int idx = tid; idx < WINW * 64; idx += 128)
    cw[idx >> 6][idx & 63] = conv_w[idx];

  for (int idx = tid; idx < (CT + 6) * 16; idx += 128) {
    int r  = idx >> 4;
    int c4 = (idx & 15) << 2;
    long t = t0 - 3 + r;
    if (t >= 0 && t < Tt) {
      async_load_lds_b128(lds_off(&hb[r][c4]), h + (base + t) * Dd + c4);
    } else {
      *reinterpret_cast<float4*>(&hb[r][c4]) = make_float4(0.f, 0.f, 0.f, 0.f);
    }
  }
  wait_asynccnt0();
  __syncthreads();

  float alpha = 1.0f / (1.0f + expf(-gate_logit[0]));

  float accv = 0.f;
  #pragma unroll
  for (int j = 0; j < WINW; ++j) {
    #pragma unroll
    for (int c4 = 0; c4 < 16; ++c4) {
      float4 hv = *reinterpret_cast<const float4*>(&hb[tid + j][c4 * 4]);
      float4 wv = *reinterpret_cast<const float4*>(&cw[j][c4 * 4]);
      accv += hv.x * wv.x + hv.y * wv.y + hv.z * wv.z + hv.w * wv.w;
    }
  }

  long t = t0 + tid;
  float ws = 0.f;
  #pragma unroll
  for (int i = 0; i < WINW; ++i)
    ws += ssf_x[(base + t) * WINW + i] * ssf_w[i];

  float wconv = accv + conv_b[0];
  float wssf  = ws + ssf_b[0];
  float a = tanhf(alpha * wconv + (1.0f - alpha) * wssf);
  a_out[base + t] = pmask[base + t] ? a : -INFINITY;
}

// ---------------------------------------------------------------------------
// Kernel 3: softmax over T per batch; writes attn (output) and av (masked)
// ---------------------------------------------------------------------------
__global__ __launch_bounds__(1024) void softmax_kernel(
    const float* __restrict__ a_in, const unsigned char* __restrict__ pmask,
    float* __restrict__ attn_out, float* __restrict__ av_out)
{
  __shared__ float red[1024];
  const int tid = threadIdx.x;
  const long base = (long)blockIdx.x * Tt;

  float v[8];
  float m = -INFINITY;
  #pragma unroll
  for (int i = 0; i < 8; ++i) {
    v[i] = a_in[base + tid * 8 + i];
    m = fmaxf(m, v[i]);
  }
  red[tid] = m; __syncthreads();
  for (int off = 512; off > 0; off >>= 1) {
    if (tid < off) red[tid] = fmaxf(red[tid], red[tid + off]);
    __syncthreads();
  }
  m = red[0]; __syncthreads();

  float e[8];
  float s = 0.f;
  #pragma unroll
  for (int i = 0; i < 8; ++i) { e[i] = expf(v[i] - m); s += e[i]; }
  red[tid] = s; __syncthreads();
  for (int off = 512; off > 0; off >>= 1) {
    if (tid < off) red[tid] += red[tid + off];
    __syncthreads();
  }
  s = red[0];

  float inv = 1.0f / s;
  #pragma unroll
  for (int i = 0; i < 8; ++i) {
    long t = base + tid * 8 + i;
    float at = e[i] * inv;
    attn_out[t] = at;
    av_out[t] = pmask[t] ? at : -INFINITY;
  }
}

// ---------------------------------------------------------------------------
// Kernel 4: top-K(1024) of 8192 per batch (jax tie semantics), indices emitted
// directly in ascending order via ordered prefix sums. grid=8, block=1024.
// ---------------------------------------------------------------------------
__global__ __launch_bounds__(1024) void topk_kernel(
    const float* __restrict__ av, int* __restrict__ idx_out)
{
  __shared__ unsigned sG[1024];
  __shared__ unsigned sE[1024];
  __shared__ unsigned sCnt;

  const int tid = threadIdx.x;
  const long base = (long)blockIdx.x * Tt;

  unsigned key[8];
  #pragma unroll
  for (int i = 0; i < 8; ++i) {
    unsigned u = __float_as_uint(av[base + tid * 8 + i]);
    key[i] = (u & 0x80000000u) ? ~u : (u | 0x80000000u);  // order-preserving
  }

  // MSB-first radix select of the K-th largest key
  unsigned prefix = 0;
  int rem = KK;
  for (int bit = 31; bit >= 0; --bit) {
    unsigned cand = prefix | (1u << bit);
    int local = 0;
    #pragma unroll
    for (int i = 0; i < 8; ++i)
      local += ((key[i] >> bit) == (cand >> bit)) ? 1 : 0;
    if (tid == 0) sCnt = 0;
    __syncthreads();
    if (local) atomicAdd(&sCnt, (unsigned)local);
    __syncthreads();
    unsigned cnt = sCnt;
    __syncthreads();
    if (cnt >= (unsigned)rem) prefix = cand; else rem -= (int)cnt;
  }
  const unsigned thr = prefix;

  unsigned g = 0, e = 0;
  #pragma unroll
  for (int i = 0; i < 8; ++i) { g += key[i] > thr; e += key[i] == thr; }
  sG[tid] = g; sE[tid] = e; __syncthreads();

  for (int off = 1; off < 1024; off <<= 1) {
    unsigned gv = 0, ev = 0;
    if (tid >= off) { gv = sG[tid - off]; ev = sE[tid - off]; }
    __syncthreads();
    sG[tid] += gv; sE[tid] += ev;
    __syncthreads();
  }

  unsigned Ctot = sG[1023];
  unsigned kEq  = (unsigned)KK - Ctot;     // equal-to-threshold slots (lowest t first)
  unsigned gB = sG[tid] - g;
  unsigned eB = sE[tid] - e;
  int* outb = idx_out + (long)blockIdx.x * KK;

  #pragma unroll
  for (int i = 0; i < 8; ++i) {
    if (key[i] > thr) {
      unsigned sel_eq_before = (eB < kEq) ? eB : kEq;
      outb[gB + sel_eq_before] = tid * 8 + i;
      gB++;
    } else if (key[i] == thr) {
      if (eB < kEq) outb[gB + eB] = tid * 8 + i;
      eB++;
    }
  }
}

// ---------------------------------------------------------------------------
// Kernel 5: pooled[b,k,:] = X[b, idx[b,k], :]
// ---------------------------------------------------------------------------
__global__ __launch_bounds__(128) void gather_kernel(
    const float* __restrict__ X, const int* __restrict__ idx,
    float* __restrict__ pooled)
{
  const int k = blockIdx.x & (KK - 1);
  const int b = blockIdx.x >> 10;
  const long src = ((long)b * Tt + idx[b * KK + k]) * Ee;
  const long dst = ((long)b * KK + k) * Ee;
  const int tid = threadIdx.x;
  float4 v = *reinterpret_cast<const float4*>(X + src + tid * 4);
  *reinterpret_cast<float4*>(pooled + dst + tid * 4) = v;
}

// ---------------------------------------------------------------------------
extern "C" void kernel_launch(void* const* d_in, const int* in_sizes, int n_in,
                              void* d_out, int out_size, void* d_ws, size_t ws_size,
                              hipStream_t stream) {
  const float* X      = (const float*)d_in[0];
  const float* ssf_x  = (const float*)d_in[1];
  const unsigned char* pmask = (const unsigned char*)d_in[2];
  const float* W1     = (const float*)d_in[3];
  const float* b1     = (const float*)d_in[4];
  const float* ln_g   = (const float*)d_in[5];
  const float* ln_b   = (const float*)d_in[6];
  const float* conv_w = (const float*)d_in[7];
  const float* conv_b = (const float*)d_in[8];
  const float* ssf_w  = (const float*)d_in[9];
  const float* ssf_b  = (const float*)d_in[10];
  const float* gate   = (const float*)d_in[11];

  float* pooled = (float*)d_out;                               // B*K*E
  float* attn   = (float*)d_out + (size_t)Bb * KK * Ee;        // B*T

  float* h_ws  = (float*)d_ws;                                 // B*T*D floats
  float* a_ws  = h_ws + (size_t)Bb * Tt * Dd;                  // B*T
  float* av_ws = a_ws + (size_t)Bb * Tt;                       // B*T
  int*   idx_ws = (int*)(av_ws + (size_t)Bb * Tt);             // B*K

  gemm_gelu_ln_kernel<<<(Bb * Tt) / TILE_M, 256, 0, stream>>>(
      X, W1, b1, ln_g, ln_b, h_ws);
  conv_gate_kernel<<<Bb * (Tt / CT), CT, 0, stream>>>(
      h_ws, conv_w, conv_b, ssf_x, ssf_w, ssf_b, gate, pmask, a_ws);
  softmax_kernel<<<Bb, 1024, 0, stream>>>(a_ws, pmask, attn, av_ws);
  topk_kernel<<<Bb, 1024, 0, stream>>>(av_ws, idx_ws);
  gather_kernel<<<Bb * KK, 128, 0, stream>>>(X, idx_ws, pooled);
}